// GatedGraphConv_936302871052
// MI455X (gfx1250) — compile-verified
//
#include <hip/hip_runtime.h>

typedef float v2f __attribute__((ext_vector_type(2)));
typedef float v8f __attribute__((ext_vector_type(8)));

#define CH 128            // feature channels
#define LDS_STRIDE 132    // 128 + 4 pad -> conflict-free fragment reads

__device__ __forceinline__ v8f wmma_f32(v2f a, v2f b, v8f c) {
    // D = A(16x4 f32) * B(4x16 f32) + C(16x16 f32)
    return __builtin_amdgcn_wmma_f32_16x16x4_f32(false, a, false, b, (short)0, c, false, false);
}

__device__ __forceinline__ float sigmoidf(float x) { return 1.0f / (1.0f + expf(-x)); }

__device__ __forceinline__ v8f splat8(float b) {
    v8f v = {b, b, b, b, b, b, b, b};
    return v;
}

// ---------------------------------------------------------------------------
// m = state @ W_lin^T    (W: [CH][CH] row-major, torch Linear layout)
// 512 threads = 16 waves; 64 rows per block; wave w -> column-group w%8,
// row-tiles {2*(w/8), 2*(w/8)+1}: two independent WMMA chains share each B
// fragment -> 2x ILP on the f32 WMMA pipe, half the weight fetches.
// ---------------------------------------------------------------------------
__global__ __launch_bounds__(512)
void k_lin(const float* __restrict__ state, const float* __restrict__ W,
           float* __restrict__ m, int N)
{
    extern __shared__ float lds[];              // 64 x LDS_STRIDE
    const int t    = threadIdx.x;
    const int wave = t >> 5;
    const int lane = t & 31;
    const int h    = lane >> 4;
    const int l    = lane & 15;
    const int rowBase = blockIdx.x * 64;

    for (int idx = t; idx < 64 * 32; idx += 512) {
        int r  = idx >> 5;
        int c4 = (idx & 31) << 2;
        int gr = rowBase + r;
        float4 v = make_float4(0.f, 0.f, 0.f, 0.f);
        if (gr < N) v = *(const float4*)(state + (size_t)gr * CH + c4);
        *(float4*)(&lds[r * LDS_STRIDE + c4]) = v;
    }
    __syncthreads();

    const int cg  = wave & 7;
    const int rt0 = (wave >> 3) * 2;
    const int col = cg * 16 + l;
    const float* A0 = &lds[(rt0 * 16 + l) * LDS_STRIDE];
    const float* A1 = &lds[((rt0 + 1) * 16 + l) * LDS_STRIDE];
    const float* Wrow = W + (size_t)col * CH;

    v8f acc0 = splat8(0.f);
    v8f acc1 = splat8(0.f);
    #pragma unroll
    for (int k0 = 0; k0 < CH; k0 += 4) {
        v2f b  = *(const v2f*)(Wrow + k0 + 2 * h);   // B[K][N=l] = W[col][K]
        v2f a0 = *(const v2f*)(A0 + k0 + 2 * h);     // A[M=l][K=2h,2h+1]
        v2f a1 = *(const v2f*)(A1 + k0 + 2 * h);
        acc0 = wmma_f32(a0, b, acc0);
        acc1 = wmma_f32(a1, b, acc1);
    }
    #pragma unroll
    for (int j = 0; j < 8; ++j) {                    // D vgpr j -> row 8h+j
        int gr0 = rowBase + rt0 * 16 + 8 * h + j;
        int gr1 = gr0 + 16;
        if (gr0 < N) m[(size_t)gr0 * CH + col] = acc0[j];
        if (gr1 < N) m[(size_t)gr1 * CH + col] = acc1[j];
    }
}

// ---------------------------------------------------------------------------
__global__ void k_zero(float4* __restrict__ p, long n4)
{
    long i = (long)blockIdx.x * blockDim.x + threadIdx.x;
    long stride = (long)gridDim.x * blockDim.x;
    float4 z = make_float4(0.f, 0.f, 0.f, 0.f);
    for (; i < n4; i += stride) p[i] = z;
}

// ---------------------------------------------------------------------------
// propagated[dst] += m[src] : one wave per edge, float4 gather + 4 L2 atomics
// ---------------------------------------------------------------------------
__global__ __launch_bounds__(256)
void k_scatter(const float* __restrict__ m, const long long* __restrict__ ei,
               float* __restrict__ prop, int E)
{
    int wid  = (blockIdx.x * 256 + threadIdx.x) >> 5;
    int lane = threadIdx.x & 31;
    if (wid >= E) return;
    int src = (int)ei[wid];
    int dst = (int)ei[(size_t)E + wid];
    float4 v = *(const float4*)(m + (size_t)src * CH + lane * 4);
    float* pd = prop + (size_t)dst * CH + lane * 4;
    atomicAdd(pd + 0, v.x);
    atomicAdd(pd + 1, v.y);
    atomicAdd(pd + 2, v.z);
    atomicAdd(pd + 3, v.w);
}

// ---------------------------------------------------------------------------
// Fused GRUCell. 256 threads = 8 waves; 32 rows per block. Wave w owns
// columns [16*w, +16) of ALL six gate tiles for BOTH row-tiles:
// 12 independent WMMA chains, each weight fragment loaded once per block,
// r/z/n combine entirely in registers.
// ---------------------------------------------------------------------------
__global__ __launch_bounds__(256)
void k_gru(const float* __restrict__ prop, const float* __restrict__ state,
           const float* __restrict__ W_ih, const float* __restrict__ W_hh,
           const float* __restrict__ b_ih, const float* __restrict__ b_hh,
           float* __restrict__ out, int N)
{
    extern __shared__ float lds[];
    float* Ap = lds;                            // 32 x LDS_STRIDE (propagated)
    float* As = lds + 32 * LDS_STRIDE;          // 32 x LDS_STRIDE (state)

    const int t    = threadIdx.x;
    const int wave = t >> 5;                    // == column group, 0..7
    const int lane = t & 31;
    const int h    = lane >> 4;
    const int l    = lane & 15;
    const int rowBase = blockIdx.x * 32;

    for (int idx = t; idx < 2 * 32 * 32; idx += 256) {
        int tile = idx >> 10;
        int rem  = idx & 1023;
        int r  = rem >> 5;
        int c4 = (rem & 31) << 2;
        int gr = rowBase + r;
        const float* srcp = tile ? state : prop;
        float* dstp       = tile ? As    : Ap;
        float4 v = make_float4(0.f, 0.f, 0.f, 0.f);
        if (gr < N) v = *(const float4*)(srcp + (size_t)gr * CH + c4);
        *(float4*)(&dstp[r * LDS_STRIDE + c4]) = v;
    }
    __syncthreads();

    const int col = wave * 16 + l;

    const float* AP0 = &Ap[l * LDS_STRIDE];
    const float* AP1 = &Ap[(16 + l) * LDS_STRIDE];
    const float* AS0 = &As[l * LDS_STRIDE];
    const float* AS1 = &As[(16 + l) * LDS_STRIDE];
    const float* Wir = W_ih + (size_t)(0 * CH + col) * CH;
    const float* Wiz = W_ih + (size_t)(1 * CH + col) * CH;
    const float* Win = W_ih + (size_t)(2 * CH + col) * CH;
    const float* Whr = W_hh + (size_t)(0 * CH + col) * CH;
    const float* Whz = W_hh + (size_t)(1 * CH + col) * CH;
    const float* Whn = W_hh + (size_t)(2 * CH + col) * CH;

    v8f cir0 = splat8(b_ih[col]),          cir1 = cir0;
    v8f ciz0 = splat8(b_ih[CH + col]),     ciz1 = ciz0;
    v8f cin0 = splat8(b_ih[2 * CH + col]), cin1 = cin0;
    v8f chr0 = splat8(b_hh[col]),          chr1 = chr0;
    v8f chz0 = splat8(b_hh[CH + col]),     chz1 = chz0;
    v8f chn0 = splat8(b_hh[2 * CH + col]), chn1 = chn0;

    #pragma unroll 2
    for (int k0 = 0; k0 < CH; k0 += 4) {
        v2f br = *(const v2f*)(Wir + k0 + 2 * h);
        v2f bz = *(const v2f*)(Wiz + k0 + 2 * h);
        v2f bn = *(const v2f*)(Win + k0 + 2 * h);
        v2f dr = *(const v2f*)(Whr + k0 + 2 * h);
        v2f dz = *(const v2f*)(Whz + k0 + 2 * h);
        v2f dn = *(const v2f*)(Whn + k0 + 2 * h);
        v2f aP0 = *(const v2f*)(AP0 + k0 + 2 * h);
        v2f aP1 = *(const v2f*)(AP1 + k0 + 2 * h);
        v2f aS0 = *(const v2f*)(AS0 + k0 + 2 * h);
        v2f aS1 = *(const v2f*)(AS1 + k0 + 2 * h);
        cir0 = wmma_f32(aP0, br, cir0);  cir1 = wmma_f32(aP1, br, cir1);
        ciz0 = wmma_f32(aP0, bz, ciz0);  ciz1 = wmma_f32(aP1, bz, ciz1);
        cin0 = wmma_f32(aP0, bn, cin0);  cin1 = wmma_f32(aP1, bn, cin1);
        chr0 = wmma_f32(aS0, dr, chr0);  chr1 = wmma_f32(aS1, dr, chr1);
        chz0 = wmma_f32(aS0, dz, chz0);  chz1 = wmma_f32(aS1, dz, chz1);
        chn0 = wmma_f32(aS0, dn, chn0);  chn1 = wmma_f32(aS1, dn, chn1);
    }

    #pragma unroll
    for (int j = 0; j < 8; ++j) {
        // row-tile 0
        int rloc0 = 8 * h + j;
        int gr0   = rowBase + rloc0;
        float r0 = sigmoidf(cir0[j] + chr0[j]);
        float z0 = sigmoidf(ciz0[j] + chz0[j]);
        float n0 = tanhf(cin0[j] + r0 * chn0[j]);
        float h0 = As[rloc0 * LDS_STRIDE + col];
        if (gr0 < N) out[(size_t)gr0 * CH + col] = (1.0f - z0) * n0 + z0 * h0;
        // row-tile 1
        int rloc1 = 16 + 8 * h + j;
        int gr1   = rowBase + rloc1;
        float r1 = sigmoidf(cir1[j] + chr1[j]);
        float z1 = sigmoidf(ciz1[j] + chz1[j]);
        float n1 = tanhf(cin1[j] + r1 * chn1[j]);
        float h1 = As[rloc1 * LDS_STRIDE + col];
        if (gr1 < N) out[(size_t)gr1 * CH + col] = (1.0f - z1) * n1 + z1 * h1;
    }
}

// ---------------------------------------------------------------------------
extern "C" void kernel_launch(void* const* d_in, const int* in_sizes, int n_in,
                              void* d_out, int out_size, void* d_ws, size_t ws_size,
                              hipStream_t stream)
{
    const float*      x     = (const float*)d_in[0];
    const long long*  ei    = (const long long*)d_in[1];
    const float*      W_lin = (const float*)d_in[2];
    const float*      W_ih  = (const float*)d_in[3];
    const float*      W_hh  = (const float*)d_in[4];
    const float*      b_ih  = (const float*)d_in[5];
    const float*      b_hh  = (const float*)d_in[6];
    float*            outp  = (float*)d_out;

    const int N = in_sizes[0] / CH;
    const int E = in_sizes[1] / 2;

    float* m    = (float*)d_ws;
    float* prop = m + (size_t)N * CH;
    float* st   = prop + (size_t)N * CH;

    const int nbLin = (N + 63) / 64;
    const int nbGru = (N + 31) / 32;
    const size_t ldsLin = (size_t)64 * LDS_STRIDE * sizeof(float);      // 33792 B
    const size_t ldsGru = (size_t)2 * 32 * LDS_STRIDE * sizeof(float);  // 33792 B
    const long  n4      = (long)N * CH / 4;
    const int   zeroBlk = 2048;
    const int   scatBlk = (E * 32 + 255) / 256;

    for (int s = 0; s < 3; ++s) {
        const float* sin  = (s == 0) ? x : st;
        float*       sout = (s == 2) ? outp : st;   // in-place safe: blocks only touch own rows
        k_lin<<<nbLin, 512, ldsLin, stream>>>(sin, W_lin, m, N);
        k_zero<<<zeroBlk, 256, 0, stream>>>((float4*)prop, n4);
        k_scatter<<<scatBlk, 256, 0, stream>>>(m, ei, prop, E);
        k_gru<<<nbGru, 256, ldsGru, stream>>>(prop, sin, W_ih, W_hh, b_ih, b_hh, sout, N);
    }
}